// SceneContraction_54795192762843
// MI455X (gfx1250) — compile-verified
//
#include <hip/hip_runtime.h>
#include <hip/hip_bf16.h>

// SceneContraction for MI455X (gfx1250).
// HBM-bound: 96 B traffic / ~120 FLOPs per sample -> optimize data movement.
// Strategy: stage each 256-sample block (3 KB means + 9 KB covs) through LDS
// using CDNA5 async global<->LDS DMA (ASYNCcnt) so all global traffic is
// aligned b128, non-temporal (streaming, no reuse -> keep 192MB L2 clean),
// then do the per-sample 3x3 Jacobian-sandwich in registers.

#define BLOCK 256
#define MEAN_F4_PER_BLOCK 192              // 256*3 floats / 4
#define SMEAN_BYTES 3072                   // 768 floats
#define LDS_BYTES 12288                    // 3072 + 9216

__device__ __forceinline__ void async_load_b128_to_lds(unsigned int lds_off,
                                                       const void* gaddr) {
  asm volatile("global_load_async_to_lds_b128 %0, %1, off th:TH_LOAD_NT"
               :: "v"(lds_off), "v"(gaddr) : "memory");
}

__device__ __forceinline__ void async_store_b128_from_lds(void* gaddr,
                                                          unsigned int lds_off) {
  asm volatile("global_store_async_from_lds_b128 %0, %1, off th:TH_STORE_NT"
               :: "v"(gaddr), "v"(lds_off) : "memory");
}

__device__ __forceinline__ void wait_asynccnt0() {
  asm volatile("s_wait_asynccnt 0" ::: "memory");
}

__device__ __forceinline__ void wait_dscnt0() {
  asm volatile("s_wait_dscnt 0" ::: "memory");
}

// out = J * C * J^T with J = s*I + c*x x^T (J symmetric).
// Expansion: s^2*C + s*c*(x v^T + w x^T) + c^2*(x^T C x)* x x^T,
// where v = C^T x, w = C x.
__device__ __forceinline__ void contract_one(float x0, float x1, float x2,
                                             const float* __restrict__ C,
                                             float* __restrict__ m,
                                             float* __restrict__ O) {
  float r2 = x0 * x0 + x1 * x1 + x2 * x2;
  float r  = sqrtf(r2);
  if (r < 1.0f) {
    m[0] = x0; m[1] = x1; m[2] = x2;
#pragma unroll
    for (int j = 0; j < 9; ++j) O[j] = C[j];
  } else {
    float ir  = 1.0f / r;
    float s   = ir * (2.0f - ir);                  // 2/r - 1/r^2
    float c   = 2.0f * ir * ir * ir * (ir - 1.0f); // 2/r^4 - 2/r^3
    m[0] = s * x0; m[1] = s * x1; m[2] = s * x2;

    float X[3] = {x0, x1, x2};
    // v = C^T x
    float V[3] = {x0 * C[0] + x1 * C[3] + x2 * C[6],
                  x0 * C[1] + x1 * C[4] + x2 * C[7],
                  x0 * C[2] + x1 * C[5] + x2 * C[8]};
    // w = C x
    float W[3] = {C[0] * x0 + C[1] * x1 + C[2] * x2,
                  C[3] * x0 + C[4] * x1 + C[5] * x2,
                  C[6] * x0 + C[7] * x1 + C[8] * x2};
    float q = x0 * W[0] + x1 * W[1] + x2 * W[2];   // x^T C x
    float a = s * s;
    float b = s * c;
    float d = c * c * q;
#pragma unroll
    for (int i = 0; i < 3; ++i)
#pragma unroll
      for (int j = 0; j < 3; ++j)
        O[i * 3 + j] = a * C[i * 3 + j] + b * (X[i] * V[j] + W[i] * X[j]) +
                       d * X[i] * X[j];
  }
}

__global__ void __launch_bounds__(BLOCK)
scene_contraction_kernel(const float* __restrict__ mean,
                         const float* __restrict__ cov,
                         float* __restrict__ out, long long n) {
  extern __shared__ float smem[];  // [0,3072) means, [3072,12288) covs (bytes)
  const int tid = threadIdx.x;
  const long long blockStart = (long long)blockIdx.x * BLOCK;
  const bool full = (blockStart + BLOCK) <= n;
  const unsigned int lds_base = __builtin_amdgcn_groupstaticsize();

  if (full) {
    // ---- stage in: 768 aligned b128 async DMA transfers per block ----
    const char* mean_b = (const char*)mean + blockStart * 3 * sizeof(float);
    const char* cov_b  = (const char*)cov  + blockStart * 9 * sizeof(float);
#pragma unroll
    for (int k = 0; k < 3; ++k) {
      const int f = tid + k * BLOCK;  // float4 index within block chunk
      // LDS offset is f*16 in both regions (cov region starts at 3072 = 192*16)
      const unsigned int lds_off = lds_base + (unsigned int)(f * 16);
      const char* g = (f < MEAN_F4_PER_BLOCK)
                          ? (mean_b + (size_t)f * 16)
                          : (cov_b + (size_t)f * 16 - SMEAN_BYTES);
      async_load_b128_to_lds(lds_off, g);
    }
    wait_asynccnt0();
    __syncthreads();

    // ---- compute: each thread owns one sample ----
    const float x0 = smem[3 * tid + 0];
    const float x1 = smem[3 * tid + 1];
    const float x2 = smem[3 * tid + 2];
    float C[9];
#pragma unroll
    for (int j = 0; j < 9; ++j) C[j] = smem[768 + 9 * tid + j];

    float m[3], O[9];
    contract_one(x0, x1, x2, C, m, O);

    // write results back into the same LDS slots (only this thread touches them)
    smem[3 * tid + 0] = m[0];
    smem[3 * tid + 1] = m[1];
    smem[3 * tid + 2] = m[2];
#pragma unroll
    for (int j = 0; j < 9; ++j) smem[768 + 9 * tid + j] = O[j];

    wait_dscnt0();
    __syncthreads();

    // ---- stage out: aligned b128 async DMA from LDS ----
    char* outm_b = (char*)out + blockStart * 3 * sizeof(float);
    char* outc_b = (char*)out + (n * 3 + blockStart * 9) * sizeof(float);
#pragma unroll
    for (int k = 0; k < 3; ++k) {
      const int f = tid + k * BLOCK;
      const unsigned int lds_off = lds_base + (unsigned int)(f * 16);
      char* g = (f < MEAN_F4_PER_BLOCK)
                    ? (outm_b + (size_t)f * 16)
                    : (outc_b + (size_t)f * 16 - SMEAN_BYTES);
      async_store_b128_from_lds(g, lds_off);
    }
    // ASYNCcnt drains via the implicit s_wait_idle at s_endpgm.
  } else {
    // ---- generic tail path (never taken for N % 256 == 0) ----
    const long long s = blockStart + tid;
    if (s < n) {
      const float* mp = mean + s * 3;
      const float* cp = cov + s * 9;
      float C[9];
#pragma unroll
      for (int j = 0; j < 9; ++j) C[j] = cp[j];
      float m[3], O[9];
      contract_one(mp[0], mp[1], mp[2], C, m, O);
      float* om = out + s * 3;
      float* oc = out + n * 3 + s * 9;
      om[0] = m[0]; om[1] = m[1]; om[2] = m[2];
#pragma unroll
      for (int j = 0; j < 9; ++j) oc[j] = O[j];
    }
  }
}

extern "C" void kernel_launch(void* const* d_in, const int* in_sizes, int n_in,
                              void* d_out, int out_size, void* d_ws, size_t ws_size,
                              hipStream_t stream) {
  const float* mean = (const float*)d_in[0];
  const float* cov  = (const float*)d_in[1];
  float* out = (float*)d_out;
  const long long n = (long long)in_sizes[0] / 3;
  const int blocks = (int)((n + BLOCK - 1) / BLOCK);
  scene_contraction_kernel<<<blocks, BLOCK, LDS_BYTES, stream>>>(mean, cov, out, n);
}